// PatchICL_Level_18399639896152
// MI455X (gfx1250) — compile-verified
//
#include <hip/hip_runtime.h>

// Problem constants (from reference)
#define Bn      16
#define Cn      3
#define Hn      512
#define PSn     64
#define Kn      64
#define STRIDEn 16          // PS/4
#define GWn     29          // (H-PS)/STRIDE + 1
#define NCELL   (GWn*GWn)   // 841
#define INV_TEMP (1.0f/0.3f)

typedef __attribute__((ext_vector_type(2))) float v2f;
typedef __attribute__((ext_vector_type(8))) float v8f;
typedef __attribute__((ext_vector_type(4))) unsigned int u32x4;
typedef __attribute__((ext_vector_type(4))) int         i32x4;
typedef __attribute__((ext_vector_type(8))) int         i32x8;

// ---------------------------------------------------------------------------
// threefry2x32 (JAX default PRNG), key = jax.random.key(42) -> (0, 42).
// jax.random.gumbel over shape (16, 841): flat iota split in halves,
// block i gets counters (i, i+half); out0 -> first half, out1 -> second half.
// ---------------------------------------------------------------------------
__device__ __forceinline__ unsigned rotl32(unsigned x, int r) {
    return (x << r) | (x >> (32 - r));
}

__device__ __forceinline__ float gumbel_noise(int j) {
    const unsigned half = (unsigned)((Bn * NCELL) / 2);   // 6728
    unsigned c0, c1; bool hi;
    if ((unsigned)j < half) { c0 = (unsigned)j; c1 = (unsigned)j + half; hi = false; }
    else                    { c0 = (unsigned)j - half; c1 = (unsigned)j; hi = true;  }
    const unsigned k0 = 0u, k1 = 42u;
    const unsigned k2 = 0x1BD11BDAu ^ k0 ^ k1;
    unsigned x0 = c0 + k0, x1 = c1 + k1;
    #define TF_R4(a,b,cc,d) \
        x0 += x1; x1 = rotl32(x1,a); x1 ^= x0; \
        x0 += x1; x1 = rotl32(x1,b); x1 ^= x0; \
        x0 += x1; x1 = rotl32(x1,cc); x1 ^= x0; \
        x0 += x1; x1 = rotl32(x1,d); x1 ^= x0;
    TF_R4(13,15,26,6)   x0 += k1; x1 += k2 + 1u;
    TF_R4(17,29,16,24)  x0 += k2; x1 += k0 + 2u;
    TF_R4(13,15,26,6)   x0 += k0; x1 += k1 + 3u;
    TF_R4(17,29,16,24)  x0 += k1; x1 += k2 + 4u;
    TF_R4(13,15,26,6)   x0 += k2; x1 += k0 + 5u;
    #undef TF_R4
    unsigned bits = hi ? x1 : x0;
    float f = __uint_as_float((bits >> 9) | 0x3f800000u) - 1.0f;  // [0,1)
    const float tiny = 1.1754943508222875e-38f;
    float u = f * (1.0f - tiny) + tiny;                           // [tiny,1)
    return -__logf(-__logf(u));
}

// ---------------------------------------------------------------------------
// Pass 1: horizontal sliding sums. One wave per (b,row); LDS-stage the row.
// rsum[b, y, gx] = sum_{dx<64} weights[b, 0, y, gx*16+dx]
// ---------------------------------------------------------------------------
__global__ void __launch_bounds__(32)
k_rowsum(const float* __restrict__ wgt, float* __restrict__ rsum) {
    int b = blockIdx.x >> 9;
    int y = blockIdx.x & (Hn - 1);
    __shared__ float row[Hn];
    const float* src = wgt + ((size_t)b * Hn + y) * Hn;
    __builtin_prefetch(src, 0, 1);
    for (int i = threadIdx.x; i < Hn; i += 32) row[i] = src[i];
    __syncthreads();
    int gx = threadIdx.x;
    if (gx < GWn) {
        float s = 0.0f;
        #pragma unroll 8
        for (int d = 0; d < PSn; ++d) s += row[gx * STRIDEn + d];
        rsum[((size_t)b * Hn + y) * GWn + gx] = s;
    }
}

// ---------------------------------------------------------------------------
// Pass 2: vertical sums -> scores; normalize; add Gumbel; iterative top-64.
// (log_softmax subtracts a per-batch constant => dropped; top-k unchanged.)
// One 256-thread block per batch image.
// ---------------------------------------------------------------------------
__global__ void __launch_bounds__(256)
k_topk(const float* __restrict__ rsum, int* __restrict__ rws, int* __restrict__ cls) {
    __shared__ float sc[NCELL];
    __shared__ float red[256];
    __shared__ int   redi[256];
    __shared__ float sMin, sMax;
    const int b = blockIdx.x, tid = threadIdx.x;

    for (int cell = tid; cell < NCELL; cell += 256) {
        int gy = cell / GWn, gx = cell % GWn;
        float s = 0.0f;
        for (int dy = 0; dy < PSn; ++dy)
            s += rsum[((size_t)b * Hn + gy * STRIDEn + dy) * GWn + gx];
        sc[cell] = s * (1.0f / (PSn * PSn));
    }
    __syncthreads();

    // min reduction
    float lm = 3.4e38f;
    for (int cell = tid; cell < NCELL; cell += 256) lm = fminf(lm, sc[cell]);
    red[tid] = lm; __syncthreads();
    for (int s = 128; s > 0; s >>= 1) {
        if (tid < s) red[tid] = fminf(red[tid], red[tid + s]);
        __syncthreads();
    }
    if (tid == 0) sMin = red[0];
    __syncthreads();
    // max reduction
    lm = -3.4e38f;
    for (int cell = tid; cell < NCELL; cell += 256) lm = fmaxf(lm, sc[cell]);
    red[tid] = lm; __syncthreads();
    for (int s = 128; s > 0; s >>= 1) {
        if (tid < s) red[tid] = fmaxf(red[tid], red[tid + s]);
        __syncthreads();
    }
    if (tid == 0) sMax = red[0];
    __syncthreads();

    float inv = 1.0f / fmaxf(sMax - sMin, 1e-6f);
    for (int cell = tid; cell < NCELL; cell += 256) {
        float key = (sc[cell] - sMin) * inv * INV_TEMP + gumbel_noise(b * NCELL + cell);
        sc[cell] = key;
    }
    __syncthreads();

    // iterative argmax x64 (tie -> smaller index, matching top_k)
    for (int t = 0; t < Kn; ++t) {
        float bv = -3.4e38f; int bi = NCELL;
        for (int cell = tid; cell < NCELL; cell += 256) {
            float v = sc[cell];
            if (v > bv || (v == bv && cell < bi)) { bv = v; bi = cell; }
        }
        red[tid] = bv; redi[tid] = bi; __syncthreads();
        for (int s = 128; s > 0; s >>= 1) {
            if (tid < s) {
                float v2 = red[tid + s]; int i2 = redi[tid + s];
                if (v2 > red[tid] || (v2 == red[tid] && i2 < redi[tid])) {
                    red[tid] = v2; redi[tid] = i2;
                }
            }
            __syncthreads();
        }
        if (tid == 0) {
            int idx = redi[0];
            int r = (idx / GWn) * STRIDEn;     // <= 448 == H-PS, clip is a no-op
            int c = (idx % GWn) * STRIDEn;
            rws[b * Kn + t] = min(r, Hn - PSn);
            cls[b * Kn + t] = min(c, Hn - PSn);
            sc[idx] = -3.4e38f;
        }
        __syncthreads();
    }
}

// ---------------------------------------------------------------------------
// Zero the accumulators (d_out is the logit accumulator, ws holds cnt).
// ---------------------------------------------------------------------------
__global__ void __launch_bounds__(256)
k_zero(float* __restrict__ a, float* __restrict__ c, int n) {
    int i = blockIdx.x * 256 + threadIdx.x;
    if (i < n) { a[i] = 0.0f; c[i] = 0.0f; }
}

// ---------------------------------------------------------------------------
// Patch kernel: TDM DMA of the 3x64x64 f32 tile (48KB) into LDS via
// TENSOR_LOAD_TO_LDS (3D tile: x=64, y=64 stride 512, z=3 stride 512^2),
// then channel-combine via V_WMMA_F32_16X16X4_F32 and scatter-add.
//   A (16x4): row m = backbone_w (lanes<16: k=0,1 ; lanes>=16: k=2, 0)
//   B (4x16): column n = pixel n channels (lane n: c0,c1 ; lane n+16: c2, x)
//   B row k=3 is multiplied by A[m][3]==0, so its value is don't-care ->
//   all operand fetches are unconditional (no EXEC divergence in the loop).
//   D row M=0 -> lane L<16 VGPR0 = f32 logit of pixel L.
// One block per (b,k) patch; 8 waves x 16 px/WMMA x 32 rounds = 4096 pixels.
// ---------------------------------------------------------------------------
__global__ void __launch_bounds__(256)
k_scatter(const float* __restrict__ img, const float* __restrict__ bw,
          const int* __restrict__ rws, const int* __restrict__ cls,
          float* __restrict__ acc, float* __restrict__ cnt) {
    __shared__ float tile[Cn * PSn * PSn];   // [c][dy][dx], 48 KB
    const int bk = blockIdx.x;
    const int b  = bk / Kn;
    const int r  = __builtin_amdgcn_readfirstlane(rws[bk]);
    const int c  = __builtin_amdgcn_readfirstlane(cls[bk]);
    const int lane = threadIdx.x & 31;
    const int wave = threadIdx.x >> 5;
    const bool lo  = lane < 16;
    const size_t HH = (size_t)Hn * Hn;

    // --- Tensor Data Mover: DMA the whole patch into LDS (wave 0 issues) ---
    if (wave == 0) {
        unsigned long long gaddr =
            (unsigned long long)(const void*)(img + (size_t)b * Cn * HH
                                              + (size_t)r * Hn + c);
        unsigned ldsOff = (unsigned)(unsigned long long)(const void*)&tile[0];
        u32x4 g0;
        g0[0] = 1u;                                    // count=1 valid descriptor
        g0[1] = ldsOff;                                // lds_addr (bytes)
        g0[2] = (unsigned)gaddr;                       // global_addr[31:0]
        g0[3] = ((unsigned)(gaddr >> 32) & 0x01FFFFFFu) | (2u << 30); // addr[56:32] | type=2
        i32x8 g1;
        g1[0] = (int)(2u << 16);                       // data_size=4B, mask=0, flags=0
        g1[1] = (int)((unsigned)Hn << 16);             // tensor_dim0[15:0] @bits[63:48]
        g1[2] = (int)((unsigned)Hn << 16);             // dim0[31:16]=0 | tensor_dim1[15:0]
        g1[3] = (int)((unsigned)PSn << 16);            // dim1[31:16]=0 | tile_dim0=64
        g1[4] = (int)((unsigned)PSn | (unsigned)Cn << 16); // tile_dim1=64, tile_dim2=3
        g1[5] = Hn;                                    // tensor_dim0_stride = 512
        g1[6] = 0;                                     // stride0 hi | stride1[15:0] (0x40000 -> 0)
        g1[7] = (int)((unsigned)(Hn * Hn) >> 16);      // tensor_dim1_stride[47:16] = 4
        i32x4 g2;
        g2[0] = Cn;                                    // tensor_dim2 = 3
        g2[1] = 0; g2[2] = 0; g2[3] = 0;
        i32x4 g3; g3[0] = 0; g3[1] = 0; g3[2] = 0; g3[3] = 0;
        i32x8 gx = {};                                 // extra group (clang-23 6-arg form)
        __builtin_amdgcn_tensor_load_to_lds(g0, g1, g2, g3, gx, 0);
        __builtin_amdgcn_s_wait_tensorcnt(0);
    }
    __syncthreads();

    const float w0 = bw[0], w1 = bw[1], w2 = bw[2];
    v2f a;
    a.x = lo ? w0 : w2;
    a.y = lo ? w1 : 0.0f;                              // A[m][3] = 0 on hi lanes

    for (int round = 0; round < (PSn * PSn) / 128; ++round) {   // 32 rounds
        int p  = round * 128 + wave * 16 + (lane & 15);
        v2f bb;
        bb.x = tile[(lo ? 0 : 2 * PSn * PSn) + p];     // c0 (lo) / c2 (hi)
        bb.y = tile[PSn * PSn + p];                    // c1 (lo) / don't-care (hi)
        v8f d = {};
        d = __builtin_amdgcn_wmma_f32_16x16x4_f32(
                /*neg_a=*/false, a, /*neg_b=*/false, bb,
                /*c_mod=*/(short)0, d, /*reuse_a=*/false, /*reuse_b=*/false);
        if (lo) {
            int dy = p >> 6, dx = p & 63;
            size_t o = (size_t)b * HH + (size_t)(r + dy) * Hn + (c + dx);
            atomicAdd(&acc[o], d[0]);
            atomicAdd(&cnt[o], 1.0f);
        }
    }
}

// ---------------------------------------------------------------------------
// aggregated = acc / max(cnt, 1)
// ---------------------------------------------------------------------------
__global__ void __launch_bounds__(256)
k_final(float* __restrict__ out, const float* __restrict__ cnt, int n) {
    int i = blockIdx.x * 256 + threadIdx.x;
    if (i < n) out[i] = out[i] / fmaxf(cnt[i], 1.0f);
}

// ---------------------------------------------------------------------------
extern "C" void kernel_launch(void* const* d_in, const int* in_sizes, int n_in,
                              void* d_out, int out_size, void* d_ws, size_t ws_size,
                              hipStream_t stream) {
    (void)in_sizes; (void)n_in; (void)out_size; (void)ws_size;
    const float* image   = (const float*)d_in[0];   // [16,3,512,512]
    const float* weights = (const float*)d_in[1];   // [16,1,512,512]
    const float* bw      = (const float*)d_in[2];   // [3]
    float* out = (float*)d_out;                     // [16,1,512,512] accumulator + result

    // workspace layout
    float* rsum = (float*)d_ws;                                   // B*H*GW floats
    size_t rsumN = (size_t)Bn * Hn * GWn;
    float* cnt  = rsum + rsumN;                                   // B*H*H floats
    size_t cntN = (size_t)Bn * Hn * Hn;
    int* rws = (int*)(cnt + cntN);                                // B*K ints
    int* cls = rws + Bn * Kn;                                     // B*K ints

    const int n = Bn * Hn * Hn;

    k_rowsum <<<Bn * Hn,        32,  0, stream>>>(weights, rsum);
    k_topk   <<<Bn,             256, 0, stream>>>(rsum, rws, cls);
    k_zero   <<<(n + 255)/256,  256, 0, stream>>>(out, cnt, n);
    k_scatter<<<Bn * Kn,        256, 0, stream>>>(image, bw, rws, cls, out, cnt);
    k_final  <<<(n + 255)/256,  256, 0, stream>>>(out, cnt, n);
}